// SparseEncoder_38233798869530
// MI455X (gfx1250) — compile-verified
//
#include <hip/hip_runtime.h>

typedef __attribute__((ext_vector_type(2))) float v2f;
typedef __attribute__((ext_vector_type(8))) float v8f;

// ------------------------------------------------------------------
// Workspace layout (float offsets).  Activation buffers are padded with a
// 1-voxel zero halo in D/H/W: [B][D+2][H+2][W+2][C].  The whole activation
// region is zeroed every call, so halos stay zero and the conv inner loop
// needs no bounds checks.  Total 38,619,776 floats (~147.3 MiB).
// ------------------------------------------------------------------
static const long OFF_X0   = 0;          // [2,22,130,130,4]    2,974,400
static const long OFF_M0   = 2974400;    // [2,20,128,128]        655,360
static const long OFF_X1A  = 3629760;    // [2,22,130,130,16]  11,897,600
static const long OFF_X1B  = 15527360;   // [2,22,130,130,16]  11,897,600
static const long OFF_X2A  = 27424960;   // [2,12,66,66,32]     3,345,408
static const long OFF_X2B  = 30770368;   // [2,12,66,66,32]     3,345,408
static const long OFF_M1   = 34115776;   // [2,10,64,64]           81,920
static const long OFF_X3A  = 34197696;   // [2,7,34,34,64]      1,035,776
static const long OFF_X3B  = 35233472;   // [2,7,34,34,64]      1,035,776
static const long OFF_M2   = 36269248;   // [2,5,32,32]            10,240
static const long OFF_X4A  = 36279488;   // [2,5,18,18,128]       414,720
static const long OFF_X4B  = 36694208;   // [2,5,18,18,128]       414,720
static const long OFF_M3   = 37108928;   // [2,3,16,16]             1,536
static const long ZERO_N   = 37110464;   // zero everything below WREP
static const long OFF_WREP = 37110464;   // repacked weights, 1,509,312
static const long WR_11  = 0;
static const long WR_12  = 1728;
static const long WR_D1  = 8640;
static const long WR_21  = 22464;
static const long WR_22  = 50112;
static const long WR_D2  = 77760;
static const long WR_31  = 133056;
static const long WR_32  = 243648;
static const long WR_D3  = 354240;
static const long WR_41  = 575424;
static const long WR_42  = 1017792;
static const long WR_OUT = 1460160;

extern __shared__ float ldsbuf[];

// ------------------------------------------------------------------
__global__ void k_zero(float* p, long n) {
  long i = (long)blockIdx.x * blockDim.x + threadIdx.x;
  long s = (long)gridDim.x * blockDim.x;
  for (; i < n; i += s) p[i] = 0.f;
}

// scatter voxels into padded grid [2,22,130,130,4] + logical mask [2,20,128,128]
__global__ void k_scatter(const float* __restrict__ vf, const int* __restrict__ co,
                          float* __restrict__ x0, float* __restrict__ m0, int n) {
  int i = blockIdx.x * blockDim.x + threadIdx.x;
  if (i >= n) return;
  int b = co[i * 4 + 0], z = co[i * 4 + 1], y = co[i * 4 + 2], x = co[i * 4 + 3];
  long pbase = (((long)(b * 22 + z + 1) * 130 + y + 1) * 130 + x + 1);
  atomicAdd(&x0[pbase * 4 + 0], vf[i * 3 + 0]);
  atomicAdd(&x0[pbase * 4 + 1], vf[i * 3 + 1]);
  atomicAdd(&x0[pbase * 4 + 2], vf[i * 3 + 2]);
  m0[(((long)(b * 20 + z) * 128 + y) * 128 + x)] = 1.0f;
}

// (CO,CI,TAPS) -> [TAPS*CIP][CO] row-major, zero-padded over ci in [CI,CIP)
__global__ void k_repack(const float* __restrict__ w, float* __restrict__ wr,
                         int CI, int CIP, int CO, int TAPS) {
  int i = blockIdx.x * blockDim.x + threadIdx.x;
  int tot = TAPS * CIP * CO;
  if (i >= tot) return;
  int co_ = i % CO;
  int r   = i / CO;
  int ci  = r % CIP;
  int tap = r / CIP;
  wr[i] = (ci < CI) ? w[((long)co_ * CI + ci) * TAPS + tap] : 0.f;
}

__global__ void k_maskdown(const float* __restrict__ mi, float* __restrict__ mo,
                           int Din, int Hin, int Win, int Dout, int Hout, int Wout) {
  int i = blockIdx.x * blockDim.x + threadIdx.x;
  int tot = 2 * Dout * Hout * Wout;
  if (i >= tot) return;
  int wx = i % Wout; int r = i / Wout;
  int hy = r % Hout; r /= Hout;
  int dz = r % Dout; int b = r / Dout;
  float any = 0.f;
  for (int kd = 0; kd < 3; ++kd) {
    int id = dz * 2 + kd - 1; if (id < 0 || id >= Din) continue;
    for (int kh = 0; kh < 3; ++kh) {
      int ih = hy * 2 + kh - 1; if (ih < 0 || ih >= Hin) continue;
      for (int kw = 0; kw < 3; ++kw) {
        int iw = wx * 2 + kw - 1; if (iw < 0 || iw >= Win) continue;
        if (mi[((long)(b * Din + id) * Hin + ih) * Win + iw] > 0.f) any = 1.f;
      }
    }
  }
  mo[i] = any;
}

// ------------------------------------------------------------------
// Stage weight panel wr[row = 0..rows)[n0..n0+16) into LDS, swizzled as
// lds[(row>>2)*64 + nn*4 + (row&3)] so a lane's B fragment (k, k+1 for one n)
// is one aligned ds_load_b64.  Uses the CDNA5 async global->LDS path.
__device__ __forceinline__ void fill_panel(float* ldsp, const float* __restrict__ wr,
                                           int rows, int CO, int n0) {
  int panel = rows << 4;   // rows * 16
  for (int i = threadIdx.x; i < panel; i += blockDim.x) {
    int nn  = i & 15;
    int row = i >> 4;
    float* dst = ldsp + (((row >> 2) << 6) + (nn << 2) + (row & 3));
    const float* gp = wr + (long)row * CO + n0 + nn;
    unsigned loff = (unsigned)(unsigned long long)(void*)dst;  // LDS byte offset
    asm volatile("global_load_async_to_lds_b32 %0, %1, off"
                 :: "v"(loff), "v"(gp) : "memory");
  }
  asm volatile("s_wait_asynccnt 0" ::: "memory");
  __syncthreads();
}

// ------------------------------------------------------------------
// Implicit-GEMM 3x3x3 conv on zero-halo padded activations, pad=1,
// stride 1|2, folded BN + ReLU + mask.  Block = 8 waves sharing one
// 16-wide output-channel panel (n0), covering 8 consecutive M-tiles
// (16 W-positions each).  K = 27*CIP in steps of 4 with
// V_WMMA_F32_16X16X4_F32; B fragments from the LDS panel; A loads are
// unconditional aligned global_load_b64 (halo supplies zeros).
// A frag (16x4): lane m=l&15, VGPR0/1 = K=kb,kb+1 with kb=2*(l>=16).
// C/D: acc[g] = D[m=g+8*(l>=16)][n=l&15].
// ------------------------------------------------------------------
__global__ void k_conv3d_wmma(const float* __restrict__ x, const float* __restrict__ wr,
                              const float* __restrict__ sc, const float* __restrict__ bi,
                              const float* __restrict__ mk, float* __restrict__ y,
                              int Din, int Hin, int Win,     // logical input dims
                              int Dout, int Hout, int Wout,  // logical output dims
                              int CIP, int CO, int stride, int ntn) {
  const int KC4 = CIP >> 2;
  const int n0  = (blockIdx.x % ntn) << 4;
  const int mb  = blockIdx.x / ntn;
  fill_panel(ldsbuf, wr, 27 * CIP, CO, n0);

  const int wave = threadIdx.x >> 5;
  const int lane = threadIdx.x & 31;
  const int wt = Wout >> 4;
  int t = mb * 8 + wave;
  const int w0 = (t % wt) << 4; t /= wt;
  const int h  = t % Hout; t /= Hout;
  const int d  = t % Dout;
  const int b  = t / Dout;

  const int PH = Hin + 2, PW = Win + 2, PD = Din + 2;
  const int m   = lane & 15;
  const int nn  = lane & 15;
  const int kb  = (lane >> 4) << 1;
  const int pw0 = (w0 + m) * stride;     // physical w index of kw=0 tap

  v8f acc = {0.f, 0.f, 0.f, 0.f, 0.f, 0.f, 0.f, 0.f};
  int tap = 0;
  for (int kd = 0; kd < 3; ++kd) {
    const int id = d * stride + kd;      // physical d index (halo-shifted)
    for (int kh = 0; kh < 3; ++kh) {
      const int ih = h * stride + kh;    // physical h index
      const float* rowp = x + (((long)(b * PD + id) * PH + ih) * PW + pw0) * CIP + kb;
      for (int kw = 0; kw < 3; ++kw, ++tap) {
        const float* px = rowp + (long)kw * CIP;               // 8B aligned
        const float* pb = ldsbuf + tap * KC4 * 64 + (nn << 2) + kb;
#pragma unroll 4
        for (int kc = 0; kc < KC4; ++kc) {
          v2f a  = *(const v2f*)(px + (kc << 2));              // global_load_b64
          v2f bf = *(const v2f*)(pb + (kc << 6));              // ds_load_b64
          acc = __builtin_amdgcn_wmma_f32_16x16x4_f32(false, a, false, bf,
                                                      (short)0, acc, false, false);
        }
      }
    }
  }

  const float s_ = sc[n0 + nn];
  const float b_ = bi[n0 + nn];
  const long obm = (((long)(b * Dout + d) * Hout + h) * Wout + w0);          // mask (logical)
  const int POH = Hout + 2, POW = Wout + 2, POD = Dout + 2;
  const long obp = (((long)(b * POD + d + 1) * POH + h + 1) * POW + w0 + 1); // y (padded)
  const int mrow = (lane >> 4) << 3;
#pragma unroll
  for (int g = 0; g < 8; ++g) {
    int mm = g + mrow;
    float mv = mk[obm + mm];
    float vo = acc[g] * s_ + b_;
    vo = vo > 0.f ? vo : 0.f;
    y[(obp + mm) * CO + n0 + nn] = vo * mv;
  }
}

// 1x1 output conv: x4 padded [2,5,18,18,128] gathered as K=384 (j = c*3 + d),
// out NCHW [2,128,16,16], folded BN + ReLU, no mask.
// 32 blocks: blockIdx & 7 -> n-tile, >>3 -> macro M block of 8 (b,h) rows.
__global__ void k_outconv_wmma(const float* __restrict__ x4, const float* __restrict__ wr,
                               const float* __restrict__ sc, const float* __restrict__ bi,
                               float* __restrict__ out) {
  const int n0 = (blockIdx.x & 7) << 4;
  const int mb = blockIdx.x >> 3;
  fill_panel(ldsbuf, wr, 384, 128, n0);

  const int wave = threadIdx.x >> 5;
  const int lane = threadIdx.x & 31;
  const int mt = mb * 8 + wave;      // 0..31 -> (b,h)
  const int b  = mt >> 4;
  const int h  = mt & 15;
  const int w  = lane & 15;          // A-row m
  const int nn = lane & 15;
  const int kb = (lane >> 4) << 1;

  v8f acc = {0.f, 0.f, 0.f, 0.f, 0.f, 0.f, 0.f, 0.f};
  const float* pb = ldsbuf + (nn << 2) + kb;
#pragma unroll 4
  for (int kc = 0; kc < 96; ++kc) {
    int j0 = kc * 4 + kb;
    int j1 = j0 + 1;
    int c0 = j0 / 3, d0 = j0 - c0 * 3;
    int c1 = j1 / 3, d1 = j1 - c1 * 3;
    v2f a, bf;
    a.x  = x4[((((long)(b * 5 + d0 + 1) * 18 + h + 1) * 18 + w + 1) << 7) + c0];
    a.y  = x4[((((long)(b * 5 + d1 + 1) * 18 + h + 1) * 18 + w + 1) << 7) + c1];
    bf   = *(const v2f*)(pb + (kc << 6));
    acc = __builtin_amdgcn_wmma_f32_16x16x4_f32(false, a, false, bf,
                                                (short)0, acc, false, false);
  }
  const float s_ = sc[n0 + nn];
  const float b_ = bi[n0 + nn];
  const int mrow = (lane >> 4) << 3;
#pragma unroll
  for (int g = 0; g < 8; ++g) {
    int ww = g + mrow;
    float vo = acc[g] * s_ + b_;
    vo = vo > 0.f ? vo : 0.f;
    out[(((long)b * 128 + (n0 + nn)) * 16 + h) * 16 + ww] = vo;
  }
}

// ------------------------------------------------------------------
static inline void conv(const float* x, const float* wr, const float* s, const float* bb,
                        const float* mk, float* y,
                        int Din, int Hin, int Win, int Dout, int Hout, int Wout,
                        int CIP, int CO, int stride, hipStream_t st) {
  int mtiles = 2 * Dout * Hout * (Wout / 16);   // divisible by 8 for all layers
  int ntn = CO / 16;
  int blocks = (mtiles / 8) * ntn;
  size_t shmem = (size_t)27 * CIP * 16 * 4;
  k_conv3d_wmma<<<blocks, 256, shmem, st>>>(x, wr, s, bb, mk, y, Din, Hin, Win,
                                            Dout, Hout, Wout, CIP, CO, stride, ntn);
}

static inline void repack(const float* w, float* wr, int CI, int CIP, int CO, int TAPS,
                          hipStream_t st) {
  int tot = TAPS * CIP * CO;
  k_repack<<<(tot + 255) / 256, 256, 0, st>>>(w, wr, CI, CIP, CO, TAPS);
}

extern "C" void kernel_launch(void* const* d_in, const int* in_sizes, int n_in,
                              void* d_out, int out_size, void* d_ws, size_t ws_size,
                              hipStream_t stream) {
  const float* vf    = (const float*)d_in[0];
  const int*   coors = (const int*)d_in[1];
  // params in setup_inputs() insertion order, each (w, s, b):
  const float* p[36];
  for (int i = 0; i < 36; ++i) p[i] = (const float*)d_in[2 + i];
  // 0-2:conv1_1  3-5:conv1_2  6-8:down1  9-11:conv2_1  12-14:conv2_2
  // 15-17:down2  18-20:conv3_1 21-23:conv3_2 24-26:down3
  // 27-29:conv4_1 30-32:conv4_2 33-35:out
  int N = in_sizes[1] / 4;

  float* ws = (float*)d_ws;
  float* X0 = ws + OFF_X0;  float* M0 = ws + OFF_M0;
  float* X1A = ws + OFF_X1A; float* X1B = ws + OFF_X1B;
  float* X2A = ws + OFF_X2A; float* X2B = ws + OFF_X2B; float* M1 = ws + OFF_M1;
  float* X3A = ws + OFF_X3A; float* X3B = ws + OFF_X3B; float* M2 = ws + OFF_M2;
  float* X4A = ws + OFF_X4A; float* X4B = ws + OFF_X4B; float* M3 = ws + OFF_M3;
  float* WR = ws + OFF_WREP;

  // 1) zero all activations + masks (halos must be zero; poison-safe every call)
  k_zero<<<4096, 256, 0, stream>>>(ws, ZERO_N);
  // 2) scatter voxels
  k_scatter<<<(N + 255) / 256, 256, 0, stream>>>(vf, coors, X0, M0, N);
  // 3) repack weights -> [tap*cip + ci][co]
  repack(p[0],  WR + WR_11, 3,   4,   16,  27, stream);
  repack(p[3],  WR + WR_12, 16,  16,  16,  27, stream);
  repack(p[6],  WR + WR_D1, 16,  16,  32,  27, stream);
  repack(p[9],  WR + WR_21, 32,  32,  32,  27, stream);
  repack(p[12], WR + WR_22, 32,  32,  32,  27, stream);
  repack(p[15], WR + WR_D2, 32,  32,  64,  27, stream);
  repack(p[18], WR + WR_31, 64,  64,  64,  27, stream);
  repack(p[21], WR + WR_32, 64,  64,  64,  27, stream);
  repack(p[24], WR + WR_D3, 64,  64,  128, 27, stream);
  repack(p[27], WR + WR_41, 128, 128, 128, 27, stream);
  repack(p[30], WR + WR_42, 128, 128, 128, 27, stream);
  repack(p[33], WR + WR_OUT, 384, 384, 128, 1, stream);
  // 4) downsample masks (logical, with bounds checks)
  k_maskdown<<<(2 * 10 * 64 * 64 + 255) / 256, 256, 0, stream>>>(M0, M1, 20, 128, 128, 10, 64, 64);
  k_maskdown<<<(2 * 5 * 32 * 32 + 255) / 256, 256, 0, stream>>>(M1, M2, 10, 64, 64, 5, 32, 32);
  k_maskdown<<<(2 * 3 * 16 * 16 + 255) / 256, 256, 0, stream>>>(M2, M3, 5, 32, 32, 3, 16, 16);
  // 5) conv chain (all buffers halo-padded)
  conv(X0,  WR + WR_11, p[1],  p[2],  M0, X1A, 20, 128, 128, 20, 128, 128, 4,   16,  1, stream);
  conv(X1A, WR + WR_12, p[4],  p[5],  M0, X1B, 20, 128, 128, 20, 128, 128, 16,  16,  1, stream);
  conv(X1B, WR + WR_D1, p[7],  p[8],  M1, X2A, 20, 128, 128, 10, 64,  64,  16,  32,  2, stream);
  conv(X2A, WR + WR_21, p[10], p[11], M1, X2B, 10, 64,  64,  10, 64,  64,  32,  32,  1, stream);
  conv(X2B, WR + WR_22, p[13], p[14], M1, X2A, 10, 64,  64,  10, 64,  64,  32,  32,  1, stream);
  conv(X2A, WR + WR_D2, p[16], p[17], M2, X3A, 10, 64,  64,  5,  32,  32,  32,  64,  2, stream);
  conv(X3A, WR + WR_31, p[19], p[20], M2, X3B, 5,  32,  32,  5,  32,  32,  64,  64,  1, stream);
  conv(X3B, WR + WR_32, p[22], p[23], M2, X3A, 5,  32,  32,  5,  32,  32,  64,  64,  1, stream);
  conv(X3A, WR + WR_D3, p[25], p[26], M3, X4A, 5,  32,  32,  3,  16,  16,  64,  128, 2, stream);
  conv(X4A, WR + WR_41, p[28], p[29], M3, X4B, 3,  16,  16,  3,  16,  16,  128, 128, 1, stream);
  conv(X4B, WR + WR_42, p[31], p[32], M3, X4A, 3,  16,  16,  3,  16,  16,  128, 128, 1, stream);
  // 6) 1x1 output conv -> d_out [2,128,16,16]
  k_outconv_wmma<<<32, 256, 24576, stream>>>(X4A, WR + WR_OUT, p[34], p[35],
                                             (float*)d_out);
}